// TransformerBlock_66219805769966
// MI455X (gfx1250) — compile-verified
//
#include <hip/hip_runtime.h>
#include <math.h>

typedef float v2f __attribute__((ext_vector_type(2)));
typedef float v8f __attribute__((ext_vector_type(8)));

#define HW 16384      // 128*128 pixels
#define IMG 128

// -------------------------------------------------------------------------
// LayerNorm over channel dim (C=128) per pixel. NCHW in -> pixel-major [HW][128] out.
// -------------------------------------------------------------------------
__global__ void ln_chan_kernel(const float* __restrict__ x,
                               const float* __restrict__ w,
                               const float* __restrict__ b,
                               float* __restrict__ y_hwc) {
  int pix = blockIdx.x * blockDim.x + threadIdx.x;
  if (pix >= HW) return;
  float s = 0.f, s2 = 0.f;
  for (int c = 0; c < 128; ++c) {
    float v = x[c * HW + pix];
    s += v; s2 += v * v;
  }
  float mu   = s * (1.0f / 128.0f);
  float var  = s2 * (1.0f / 128.0f) - mu * mu;
  float rstd = rsqrtf(var + 1e-5f);
  for (int c = 0; c < 128; ++c) {
    float v = x[c * HW + pix];
    y_hwc[pix * 128 + c] = (v - mu) * rstd * w[c] + b[c];
  }
}

// -------------------------------------------------------------------------
// 1x1 conv as GEMM on the WMMA pipe, full fp32 precision:
//   C[m][n] = sum_k A[m][k] * Bh[n][k]  (+ res[m][n])
// A: weights [M][K] row-major. Bh: activations pixel-major [HW][K].
// C: NCHW [M][HW].
//
// Block = 4 waves. Each wave: one 16-row M-tile x four 16-col N-tiles.
// The 64-pixel B panel is staged into LDS per 128-column K-chunk and shared
// by all 4 waves; each A fragment is reused across 4 WMMAs.
// Out-of-range M rows are CLAMPED (not predicated): they only feed C rows
// that the store guard discards, so EXEC stays all-1s (WMMA requirement)
// with no divergent branches in the hot loop.
//
// f32 16x16x4 VGPR layouts (ISA 7.12.2):
//   A (16x4): lanes 0-15 = M rows {K=0 in v0, K=1 in v1}; lanes 16-31 {K=2,K=3}
//   B (4x16): v0 = rows K=0 (lanes 0-15) / K=2 (lanes 16-31); v1 = K=1 / K=3
//   C (16x16): VGPR v = row M=v (lanes 0-15, N=lane) and M=v+8 (lanes 16-31)
// -------------------------------------------------------------------------
#define LDS_STRIDE 132   // 128 cols + pad (bank-conflict avoidance)

__global__ void gemm_wmma_f32(const float* __restrict__ A,
                              const float* __restrict__ Bh,
                              const float* __restrict__ res,
                              float* __restrict__ C,
                              int M, int K) {
  __shared__ float lds[64 * LDS_STRIDE];   // 33.8 KB

  const int lane  = threadIdx.x;           // 0..31
  const int tid   = threadIdx.y * 32 + lane;
  const int mh    = lane & 15;
  const int khalf = lane >> 4;             // 0 -> K{0,1}, 1 -> K{2,3}

  const int MT = (M + 15) >> 4;
  int mt = blockIdx.x * 4 + threadIdx.y;
  if (mt >= MT) mt = MT - 1;               // clamp (no early return: barrier-safe)
  const int m0     = mt * 16;
  const int n0base = blockIdx.y * 64;

  int am = m0 + mh;
  if (am >= M) am = M - 1;                 // clamped A row; result rows >= M never stored
  const float* Arow = A + (size_t)am * K;

  v8f acc0 = {0.f,0.f,0.f,0.f,0.f,0.f,0.f,0.f};
  v8f acc1 = {0.f,0.f,0.f,0.f,0.f,0.f,0.f,0.f};
  v8f acc2 = {0.f,0.f,0.f,0.f,0.f,0.f,0.f,0.f};
  v8f acc3 = {0.f,0.f,0.f,0.f,0.f,0.f,0.f,0.f};

  for (int k0 = 0; k0 < K; k0 += 128) {
    int kc = K - k0;
    if (kc > 128) kc = 128;

    // cooperative stage: B panel [64 pixels][kc] -> LDS
    for (int e = tid; e < 64 * kc; e += 128) {
      int r  = e / kc;
      int cc = e - r * kc;
      lds[r * LDS_STRIDE + cc] = Bh[(size_t)(n0base + r) * K + k0 + cc];
    }
    __syncthreads();

    for (int kk = 0; kk < kc; kk += 4) {
      int kcol = kk + khalf * 2;
      v2f a  = *(const v2f*)(Arow + k0 + kcol);
      v2f b0 = *(const v2f*)(&lds[( 0 + mh) * LDS_STRIDE + kcol]);
      v2f b1 = *(const v2f*)(&lds[(16 + mh) * LDS_STRIDE + kcol]);
      v2f b2 = *(const v2f*)(&lds[(32 + mh) * LDS_STRIDE + kcol]);
      v2f b3 = *(const v2f*)(&lds[(48 + mh) * LDS_STRIDE + kcol]);
      acc0 = __builtin_amdgcn_wmma_f32_16x16x4_f32(false, a, false, b0, (short)0, acc0, false, false);
      acc1 = __builtin_amdgcn_wmma_f32_16x16x4_f32(false, a, false, b1, (short)0, acc1, false, false);
      acc2 = __builtin_amdgcn_wmma_f32_16x16x4_f32(false, a, false, b2, (short)0, acc2, false, false);
      acc3 = __builtin_amdgcn_wmma_f32_16x16x4_f32(false, a, false, b3, (short)0, acc3, false, false);
    }
    __syncthreads();
  }

#pragma unroll
  for (int j = 0; j < 4; ++j) {
    const v8f& acc = (j == 0) ? acc0 : (j == 1) ? acc1 : (j == 2) ? acc2 : acc3;
    int n = n0base + j * 16 + mh;
#pragma unroll
    for (int v = 0; v < 8; ++v) {
      int m = m0 + khalf * 8 + v;
      if (m < M) {
        float val = acc[v];
        if (res) val += res[(size_t)m * HW + n];
        C[(size_t)m * HW + n] = val;
      }
    }
  }
}

// -------------------------------------------------------------------------
// Depthwise 3x3, stride 1, zero pad 1, NCHW in/out. w: [C][1][3][3].
// -------------------------------------------------------------------------
__global__ void dw3_nchw_kernel(const float* __restrict__ in,
                                const float* __restrict__ w,
                                float* __restrict__ out, int C) {
  int idx = blockIdx.x * blockDim.x + threadIdx.x;
  if (idx >= C * HW) return;
  int c   = idx >> 14;
  int pix = idx & (HW - 1);
  int py = pix >> 7, px = pix & 127;
  const float* ip = in + (size_t)c * HW;
  const float* wp = w + c * 9;
  float s = 0.f;
#pragma unroll
  for (int dy = 0; dy < 3; ++dy) {
    int yy = py + dy - 1;
    if (yy < 0 || yy >= IMG) continue;
#pragma unroll
    for (int dx = 0; dx < 3; ++dx) {
      int xx = px + dx - 1;
      if (xx < 0 || xx >= IMG) continue;
      s += wp[dy * 3 + dx] * ip[yy * IMG + xx];
    }
  }
  out[idx] = s;
}

// -------------------------------------------------------------------------
// NATTEN 7x7 neighborhood attention. qkv buffer is physically NCHW
// (384 x HW) but indexed flat as [pix*384 + s*128 + head*32 + d] to
// reproduce the reference's raw-reshape semantics exactly.
// heads 0,1: dilation 1 / rpb0 ; heads 2,3: dilation 2 / rpb1.
// Output written pixel-major [HW][128], channel = head*32 + d.
// -------------------------------------------------------------------------
__device__ __forceinline__ void natten_start(int i, int dil, int& ni, int& pi) {
  const int L = IMG, K = 7, r = 3;
  if (dil <= 1) {
    int t = i - r;
    ni = min(max(t, 0), L - K);
    pi = r + (i < r ? r - i : 0) + (i + r >= L ? L - i - 1 - r : 0);
  } else {
    int a = (L / dil) * dil;
    int b = L - a;
    int imodd = i % dil;
    int end = (imodd < b) ? (L - b + imodd - 2 * r * dil) : (a + imodd - K * dil);
    ni = (i - r * dil < 0) ? imodd : ((i + r * dil >= L) ? end : (i - r * dil));
    pi = (i - r * dil < 0) ? (K - 1 - i / dil)
                           : ((i + r * dil >= L) ? (L - i - 1) / dil : r);
  }
}

__global__ void natten_kernel(const float* __restrict__ qkv,   // flat 384*HW
                              const float* __restrict__ rpb0,  // (2,13,13)
                              const float* __restrict__ rpb1,  // (2,13,13)
                              float* __restrict__ o_hwc) {     // [HW][128]
  int idx = blockIdx.x * blockDim.x + threadIdx.x;
  if (idx >= HW * 4) return;
  int head = idx & 3;
  int pix  = idx >> 2;
  int px = pix & 127, py = pix >> 7;
  int dil = (head < 2) ? 1 : 2;
  const float* rpb = (head < 2) ? (rpb0 + head * 169) : (rpb1 + (head - 2) * 169);
  int niH, piH, niW, piW;
  natten_start(py, dil, niH, piH);
  natten_start(px, dil, niW, piW);

  const float scale = 0.17677669529663687f;  // 32^-0.5
  float q[32];
  const float* qp = qkv + (size_t)pix * 384 + head * 32;
#pragma unroll
  for (int i = 0; i < 32; ++i) q[i] = qp[i] * scale;

  float mx = -INFINITY, l = 0.f;
  float acc[32];
#pragma unroll
  for (int i = 0; i < 32; ++i) acc[i] = 0.f;

  for (int k1 = 0; k1 < 7; ++k1) {
    int ky = niH + k1 * dil;
    for (int k2 = 0; k2 < 7; ++k2) {
      int kx = niW + k2 * dil;
      const float* kp = qkv + (size_t)(ky * IMG + kx) * 384 + 128 + head * 32;
      float s = 0.f;
#pragma unroll
      for (int i = 0; i < 32; ++i) s += q[i] * kp[i];
      s += rpb[(piH + k1) * 13 + (piW + k2)];
      float nm = fmaxf(mx, s);
      float ew = expf(s - nm);
      float f  = expf(mx - nm);
      l = l * f + ew;
      const float* vp = kp + 128;  // v is 128 channels past k in the flat view
#pragma unroll
      for (int i = 0; i < 32; ++i) acc[i] = acc[i] * f + ew * vp[i];
      mx = nm;
    }
  }
  float inv = 1.f / l;
  float* op = o_hwc + (size_t)pix * 128 + head * 32;
#pragma unroll
  for (int i = 0; i < 32; ++i) op[i] = acc[i] * inv;
}

// -------------------------------------------------------------------------
// FFN: fused depthwise 3x3 (both halves) + exact GELU gating.
// t: NCHW [680][HW]; out: pixel-major [HW][340] for the final WMMA GEMM.
// -------------------------------------------------------------------------
__global__ void ffn_dwgate_kernel(const float* __restrict__ t,
                                  const float* __restrict__ w,   // (680,1,3,3)
                                  float* __restrict__ g_hwc) {   // [HW][340]
  int idx = blockIdx.x * blockDim.x + threadIdx.x;
  if (idx >= 340 * HW) return;
  int c   = idx >> 14;
  int pix = idx & (HW - 1);
  int py = pix >> 7, px = pix & 127;
  const float* i1 = t + (size_t)c * HW;
  const float* i2 = t + (size_t)(c + 340) * HW;
  const float* w1 = w + c * 9;
  const float* w2 = w + (c + 340) * 9;
  float s1 = 0.f, s2 = 0.f;
#pragma unroll
  for (int dy = 0; dy < 3; ++dy) {
    int yy = py + dy - 1;
    if (yy < 0 || yy >= IMG) continue;
#pragma unroll
    for (int dx = 0; dx < 3; ++dx) {
      int xx = px + dx - 1;
      if (xx < 0 || xx >= IMG) continue;
      float wa = w1[dy * 3 + dx], wb = w2[dy * 3 + dx];
      s1 += wa * i1[yy * IMG + xx];
      s2 += wb * i2[yy * IMG + xx];
    }
  }
  float g = 0.5f * s1 * (1.f + erff(s1 * 0.7071067811865476f)); // exact GELU
  g_hwc[(size_t)pix * 340 + c] = g * s2;
}

// -------------------------------------------------------------------------
// Orchestration. Workspace layout (floats):
//   [0,        2M)  x1        (residual after attention)
//   [2M,       4M)  y_hwc     (LN output, reused for LN1/LN2)
//   [4M,    9.57M)  o_hwc (2M) then g_hwc (5.57M)
//   [9.57M, 22.15M) qkv_pre(6M)+qkv_dw(6M), later reused as ffn_pre(11.1M)
// Total ~89.4 MB of d_ws.
// -------------------------------------------------------------------------
extern "C" void kernel_launch(void* const* d_in, const int* in_sizes, int n_in,
                              void* d_out, int out_size, void* d_ws, size_t ws_size,
                              hipStream_t stream) {
  const float* x         = (const float*)d_in[0];
  const float* ln1_w     = (const float*)d_in[1];
  const float* ln1_b     = (const float*)d_in[2];
  const float* qkv_w1    = (const float*)d_in[3];
  const float* qkv_dw_w  = (const float*)d_in[4];
  const float* rpb0      = (const float*)d_in[5];
  const float* rpb1      = (const float*)d_in[6];
  const float* proj_w    = (const float*)d_in[7];
  const float* ln2_w     = (const float*)d_in[8];
  const float* ln2_b     = (const float*)d_in[9];
  const float* ffn_in_w  = (const float*)d_in[10];
  const float* ffn_dw_w  = (const float*)d_in[11];
  const float* ffn_out_w = (const float*)d_in[12];
  float* out = (float*)d_out;
  float* ws  = (float*)d_ws;

  float* x1      = ws;                 // 2,097,152
  float* ybuf    = ws + 2097152;       // 2,097,152
  float* og      = ws + 4194304;       // up to 5,570,560 (o_hwc then g_hwc)
  float* big     = ws + 9764864;       // 12,582,912 region
  float* qkv_pre = big;                // 6,291,456
  float* qkv_dwb = big + 6291456;      // 6,291,456
  float* ffn_pre = big;                // 11,141,120 (reuses qkv region)

  dim3 b256(256);
  dim3 gblk(32, 4);       // 4 waves; each wave: 16 M-rows x 64 pixels
  dim3 ggrid_n(0, HW / 64);

  // ---- attention branch ----
  ln_chan_kernel<<<dim3(HW / 256), b256, 0, stream>>>(x, ln1_w, ln1_b, ybuf);

  { // qkv 1x1: M=384, K=128
    int MT = (384 + 15) / 16;
    gemm_wmma_f32<<<dim3((MT + 3) / 4, HW / 64), gblk, 0, stream>>>(
        qkv_w1, ybuf, nullptr, qkv_pre, 384, 128);
  }
  dw3_nchw_kernel<<<dim3((384 * HW) / 256), b256, 0, stream>>>(
      qkv_pre, qkv_dw_w, qkv_dwb, 384);
  natten_kernel<<<dim3((HW * 4) / 256), b256, 0, stream>>>(
      qkv_dwb, rpb0, rpb1, og);
  { // proj 1x1 + residual(x) -> x1 : M=128, K=128
    int MT = (128 + 15) / 16;
    gemm_wmma_f32<<<dim3((MT + 3) / 4, HW / 64), gblk, 0, stream>>>(
        proj_w, og, x, x1, 128, 128);
  }

  // ---- gated-dconv FFN branch ----
  ln_chan_kernel<<<dim3(HW / 256), b256, 0, stream>>>(x1, ln2_w, ln2_b, ybuf);
  { // ffn_in 1x1: M=680, K=128
    int MT = (680 + 15) / 16;
    gemm_wmma_f32<<<dim3((MT + 3) / 4, HW / 64), gblk, 0, stream>>>(
        ffn_in_w, ybuf, nullptr, ffn_pre, 680, 128);
  }
  ffn_dwgate_kernel<<<dim3((340 * HW) / 256), b256, 0, stream>>>(
      ffn_pre, ffn_dw_w, og);
  { // ffn_out 1x1 + residual(x1) -> out : M=128, K=340
    int MT = (128 + 15) / 16;
    gemm_wmma_f32<<<dim3((MT + 3) / 4, HW / 64), gblk, 0, stream>>>(
        ffn_out_w, og, x1, out, 128, 340);
  }
}